// EdgeClassification_70093866271399
// MI455X (gfx1250) — compile-verified
//
#include <hip/hip_runtime.h>

// N=1024 nodes, H=256 hidden, C=10 classes (fixed by setup_inputs()).
// out[i,j,c] = (emb[i]+emb[j])·W[c] + b[c] = Q[i,c] + Q[j,c] + b[c],
// with Q = emb @ W^T  ([1024,10], padded to [1024,16] in workspace).
// Roofline: output write 41.9 MB / 23.3 TB/s ~ 1.8 us is the floor; the GEMM
// stage is 2.6 MFLOP and exists only to produce Q exactly (fp32 WMMA).

typedef __attribute__((ext_vector_type(2))) float v2f;
typedef __attribute__((ext_vector_type(8))) float v8f;

// ---------------------------------------------------------------------------
// Phase 1: Q[i,c] = sum_h emb[i,h] * W[c,h], via V_WMMA_F32_16X16X4_F32.
// One wave (32 threads) per 16-row tile of emb. B = W^T; columns 10..15 of
// the D tile are never consumed downstream, so the B column address is simply
// clamped to 9 (branch-free, EXEC stays all-1s through the whole loop).
// ---------------------------------------------------------------------------
__global__ __launch_bounds__(32)
void edge_cls_gemm_wmma(const float* __restrict__ emb,
                        const float* __restrict__ W,
                        float* __restrict__ Qpad /* [1024][16] */) {
  const int lane   = threadIdx.x;   // wave32
  const int half   = lane >> 4;     // 0 -> K pair {0,1}, 1 -> K pair {2,3}
  const int lanelo = lane & 15;
  const int m0     = blockIdx.x * 16;

  const int arow = m0 + lanelo;                  // A-matrix row this lane holds
  const int bcol = (lanelo < 10) ? lanelo : 9;   // clamped B column (class c)

  const float* __restrict__ aptr = emb + arow * 256 + half * 2;
  const float* __restrict__ bptr = W   + bcol * 256 + half * 2;

  v8f acc = {};
#pragma unroll 8
  for (int k0 = 0; k0 < 256; k0 += 4) {
    // A 16x4 f32 layout: lanes 0-15 hold (M=lane,   K=k0..k0+1),
    //                    lanes 16-31 hold (M=lane-16,K=k0+2..k0+3).
    v2f a  = *(const v2f*)(aptr + k0);
    // B 4x16 f32 layout (mirror of A): lane%16 = N column, same K split.
    v2f bb = *(const v2f*)(bptr + k0);
    acc = __builtin_amdgcn_wmma_f32_16x16x4_f32(
        /*neg_a=*/false, a, /*neg_b=*/false, bb,
        /*c_mod=*/(short)0, acc, /*reuse_a=*/false, /*reuse_b=*/false);
  }

  // D layout: VGPR r -> lanes 0-15: (M=r, N=lane); lanes 16-31: (M=r+8, N=lane-16)
#pragma unroll
  for (int r = 0; r < 8; ++r) {
    const int row = m0 + r + half * 8;
    Qpad[row * 16 + lanelo] = acc[r];
  }
}

// ---------------------------------------------------------------------------
// Phase 2: out[i*N + j, c] = (Q[i,c] + b[c]) + Q[j,c].
// One block per i. Row = 1024*10 = 10240 floats = 2560 float4 (rows are
// 40960 B apart -> 16B aligned). 256 threads x 10 iterations of coalesced
// global_store_b128. Since e = 4*f has e%10 even, each float4 splits into two
// 8B-aligned pairs -> b64 loads from Qpad (hot in L0/L2) and LDS.
// ---------------------------------------------------------------------------
__global__ __launch_bounds__(256)
void edge_cls_broadcast(const float* __restrict__ Qpad,
                        const float* __restrict__ bias,
                        float* __restrict__ out) {
  const int i   = blockIdx.x;
  const int tid = threadIdx.x;

  __shared__ float sQi[16];
  if (tid < 16) {
    float v = Qpad[i * 16 + tid];
    if (tid < 10) v += bias[tid];
    sQi[tid] = v;
  }
  __syncthreads();

  float4* out4 = (float4*)(out + (size_t)i * 10240u);

#pragma unroll
  for (int it = 0; it < 10; ++it) {
    const int f = it * 256 + tid;  // float4 index in row, 0..2559
    const int e = f * 4;           // element index, 0..10239
    const int j0 = e / 10;         // const-divisor -> mul/shift
    const int c0 = e - j0 * 10;    // even: 0,2,4,6,8
    int j1, c1;
    if (c0 + 2 < 10) { j1 = j0;     c1 = c0 + 2; }
    else             { j1 = j0 + 1; c1 = c0 - 8; }

    const float2 qa = *(const float2*)(Qpad + j0 * 16 + c0);
    const float2 qb = *(const float2*)(Qpad + j1 * 16 + c1);
    const float2 sa = *(const float2*)(sQi + c0);
    const float2 sb = *(const float2*)(sQi + c1);

    float4 v;
    v.x = sa.x + qa.x;
    v.y = sa.y + qa.y;
    v.z = sb.x + qb.x;
    v.w = sb.y + qb.y;
    out4[f] = v;
  }
}

// ---------------------------------------------------------------------------
extern "C" void kernel_launch(void* const* d_in, const int* in_sizes, int n_in,
                              void* d_out, int out_size, void* d_ws, size_t ws_size,
                              hipStream_t stream) {
  const float* emb  = (const float*)d_in[0];  // [1024, 256] f32
  const float* W    = (const float*)d_in[1];  // [10, 256] f32
  const float* bias = (const float*)d_in[2];  // [10] f32
  float* Qpad = (float*)d_ws;                 // [1024][16] f32 = 64 KB scratch

  edge_cls_gemm_wmma<<<64, 32, 0, stream>>>(emb, W, Qpad);
  edge_cls_broadcast<<<1024, 256, 0, stream>>>(Qpad, bias, (float*)d_out);
}